// LNCCLoss_15298673508911
// MI455X (gfx1250) — compile-verified
//
#include <hip/hip_runtime.h>
#include <hip/hip_bf16.h>

// LNCC loss, (4,1,128,128,128) fp32, scales {32,64}, weights {0.4,0.6}, dilation 2.
// Bandwidth-bound: 67 MB input read @ 23.3 TB/s ~ 2.9 us floor. Separable box
// filter: fused streaming x-pass (reads inputs once, async global->LDS copies),
// then tap-selection-matrix matmuls (P[5x128] x M) on y/z axes via fp32 WMMA.

#define USE_ASYNC_LDS 1   // gfx1250 GLOBAL_LOAD_ASYNC_TO_LDS_B128 path (ASYNCcnt)

typedef float v2f __attribute__((ext_vector_type(2)));
typedef float v8f __attribute__((ext_vector_type(8)));

#define NB   4        // batch
#define DIM  128      // D = H = W
#define O1   5        // scale-32 outputs per dim
#define K1   32
#define S1   16
#define K2   64

#if USE_ASYNC_LDS
// Per-lane async copy of 16 bytes global -> LDS. vdst = LDS byte offset,
// vaddr = 64-bit global address (ISA 15.18.3 op 98). Tracked by ASYNCcnt.
__device__ inline void async_g2l_b128(unsigned lds_off, const float* gptr) {
  asm volatile("global_load_async_to_lds_b128 %0, %1, off"
               :: "v"(lds_off), "v"(gptr) : "memory");
}
__device__ inline void wait_asynccnt0() {
  asm volatile("s_wait_asynccnt 0x0" ::: "memory");
}
#endif

// ---------------------------------------------------------------------------
// Stage A: streaming x-reduction for BOTH scales and all 5 quantities.
// One wave per (b, z, y-tile of 32 rows). LDS rows padded to 132 floats
// (bank = (4*y + x) mod 64 -> at worst 2-way conflicts, float4-aligned rows).
// ---------------------------------------------------------------------------
__global__ __launch_bounds__(32) void lncc_stageA(const float* __restrict__ gx,
                                                  const float* __restrict__ gt,
                                                  float* __restrict__ ws1,   // [5][4][128][128][5]
                                                  float* __restrict__ ws2) { // [5][4][128][128]
  constexpr int LSTR = 132;
  extern __shared__ float lds[];               // 2 * 32 * 132 floats = 33792 B
  float* lx = lds;
  float* lt = lds + 32 * LSTR;

  const int blk = blockIdx.x;                  // ((b*128 + z)*4 + ty)
  const int ty  = blk & 3;
  const int bz  = blk >> 2;
  const int z   = bz & (DIM - 1);
  const int b   = bz >> 7;
  const int y0  = ty * 32;
  const int tid = threadIdx.x;

  const size_t gbase = (((size_t)b * DIM + z) * DIM + y0) * DIM;
  const float* gxp = gx + gbase;
  const float* gtp = gt + gbase;

  // speculative prefetch of the next y-tile (dropped if OOB) -> global_prefetch_b8
  __builtin_prefetch(gxp + 32 * DIM, 0, 1);
  __builtin_prefetch(gtp + 32 * DIM, 0, 1);

#if USE_ASYNC_LDS
  for (int i = tid; i < (32 * DIM) / 4; i += 32) {   // 1024 x b128 per array
    int r = i >> 5;              // row in tile
    int c = (i & 31) << 2;       // col (multiple of 4)
    // generic->LDS offset: LDS aperture keeps byte offset in addr[31:0]
    async_g2l_b128((unsigned)(size_t)(lx + r * LSTR + c), gxp + 4 * i);
    async_g2l_b128((unsigned)(size_t)(lt + r * LSTR + c), gtp + 4 * i);
  }
  wait_asynccnt0();
  __syncthreads();
#else
  {
    const float4* vx = (const float4*)gxp;
    const float4* vt = (const float4*)gtp;
    for (int i = tid; i < (32 * DIM) / 4; i += 32) {
      int r = i >> 5;
      int c = (i & 31) << 2;
      *(float4*)(lx + r * LSTR + c) = vx[i];
      *(float4*)(lt + r * LSTR + c) = vt[i];
    }
  }
  __syncthreads();
#endif

  const float* rx = lx + tid * LSTR;
  const float* rt = lt + tid * LSTR;
  const int y = y0 + tid;

  // ---- scale 32: taps x = 16*ox + 2k, k<32 ----
  const size_t q_str1 = (size_t)NB * DIM * DIM * O1;
  for (int ox = 0; ox < O1; ++ox) {
    float s0 = 0.f, s1 = 0.f, s2 = 0.f, s3 = 0.f, s4 = 0.f;
    const int xb = ox * S1;
#pragma unroll 8
    for (int k = 0; k < K1; ++k) {
      float a = rx[xb + 2 * k];
      float c = rt[xb + 2 * k];
      s0 += a; s1 += c; s2 += a * a; s3 += c * c; s4 += a * c;
    }
    const size_t o = ((((size_t)b * DIM) + z) * DIM + y) * O1 + ox;
    ws1[o]              = s0;
    ws1[o +     q_str1] = s1;
    ws1[o + 2 * q_str1] = s2;
    ws1[o + 3 * q_str1] = s3;
    ws1[o + 4 * q_str1] = s4;
  }

  // ---- scale 64: taps x = 2k, k<64 ----
  {
    float s0 = 0.f, s1 = 0.f, s2 = 0.f, s3 = 0.f, s4 = 0.f;
#pragma unroll 8
    for (int k = 0; k < K2; ++k) {
      float a = rx[2 * k];
      float c = rt[2 * k];
      s0 += a; s1 += c; s2 += a * a; s3 += c * c; s4 += a * c;
    }
    const size_t q_str2 = (size_t)NB * DIM * DIM;
    const size_t o = (((size_t)b * DIM) + z) * DIM + y;
    ws2[o]              = s0;
    ws2[o +     q_str2] = s1;
    ws2[o + 2 * q_str2] = s2;
    ws2[o + 3 * q_str2] = s3;
    ws2[o + 4 * q_str2] = s4;
  }
}

// ---------------------------------------------------------------------------
// Tap-selection A-tile for WMMA: A[m][k] = 1 iff source position z = 4c+k is a
// tap of output m (z - 16m in {0,2,...,62}). Layout per ISA 7.12.2 (32-bit A):
// lanes 0-15: M=lane, V0=K0,V1=K1 ; lanes 16-31: M=lane-16, V0=K2,V1=K3.
// ---------------------------------------------------------------------------
__device__ inline v2f sel_tile(int c, int lane) {
  int m  = lane & 15;
  int kb = (lane >> 4) * 2;
  int z0 = 4 * c + kb;
  int d0 = z0 - S1 * m;
  int d1 = z0 + 1 - S1 * m;
  v2f a;
  a[0] = (m < O1 && d0 >= 0 && d0 <= 62 && ((d0 & 1) == 0)) ? 1.0f : 0.0f;
  a[1] = (m < O1 && d1 >= 0 && d1 <= 62 && ((d1 & 1) == 0)) ? 1.0f : 0.0f;
  return a;
}

// ---------------------------------------------------------------------------
// Stage B: scale-32 y-reduction as D(16x16) += A(16x4) x B(4x16), fp32 WMMA.
// One wave per (q,b,z): out[oy][ox] = sum_y P[oy][y] * src[y][ox].
// B layout: lanes 0-15 hold K=0,1 (V0,V1); lanes 16-31 hold K=2,3; N = lane&15.
// B columns n>=5 are never stored from D, so we load them UNCONDITIONALLY
// (benign overread of a few floats into the next d_ws region) -> no EXEC
// divergence in the WMMA loop.
// ---------------------------------------------------------------------------
__global__ __launch_bounds__(32) void lncc_stageB(const float* __restrict__ ws1,   // [pid][128][5]
                                                  float* __restrict__ ws1b) {      // [pid][5][5]
  const int pid  = blockIdx.x;           // (q*4+b)*128 + z, 2560 total
  const int lane = threadIdx.x;
  const float* src = ws1 + (size_t)pid * (DIM * O1);

  const int n  = lane & 15;
  const int kb = (lane >> 4) * 2;
  const float* p0 = src + kb * O1 + n;   // row kb,   col n (garbage if n>=5: ok)
  const float* p1 = p0 + O1;             // row kb+1, col n

  v8f acc = {0.f, 0.f, 0.f, 0.f, 0.f, 0.f, 0.f, 0.f};
  for (int c = 0; c < DIM / 4; ++c) {
    v2f A = sel_tile(c, lane);
    v2f Bm;
    Bm[0] = p0[c * 4 * O1];
    Bm[1] = p1[c * 4 * O1];
    acc = __builtin_amdgcn_wmma_f32_16x16x4_f32(false, A, false, Bm,
                                                (short)0, acc, false, false);
  }

  // D layout: lanes 0-15: N=lane, M=r ; lanes 16-31: N=lane-16, M=r+8
  float* dst = ws1b + (size_t)pid * (O1 * O1);
  const int mbase = (lane >> 4) * 8;
  for (int r = 0; r < 8; ++r) {
    int m = mbase + r;
    if (m < O1 && n < O1) dst[m * O1 + n] = acc[r];
  }
}

// ---------------------------------------------------------------------------
// Stage C: scale-32 z-reduction, same WMMA form. One wave per (q*4+b, coltile).
// cols c25 = oy*5+ox (25 valid, two 16-wide tiles). Unconditional loads as in
// stage B (overread stays inside d_ws; those D columns are discarded).
// ---------------------------------------------------------------------------
__global__ __launch_bounds__(32) void lncc_stageC(const float* __restrict__ ws1b,  // [qb][128][25]
                                                  float* __restrict__ ws1c) {      // [qb][5][25]
  const int qb   = blockIdx.x >> 1;      // 0..19
  const int tile = blockIdx.x & 1;
  const int lane = threadIdx.x;
  const float* src = ws1b + (size_t)qb * (DIM * 25);

  const int n   = lane & 15;
  const int col = tile * 16 + n;
  const int kb  = (lane >> 4) * 2;
  const float* p0 = src + kb * 25 + col;
  const float* p1 = p0 + 25;

  v8f acc = {0.f, 0.f, 0.f, 0.f, 0.f, 0.f, 0.f, 0.f};
  for (int c = 0; c < DIM / 4; ++c) {
    v2f A = sel_tile(c, lane);
    v2f Bm;
    Bm[0] = p0[c * 4 * 25];
    Bm[1] = p1[c * 4 * 25];
    acc = __builtin_amdgcn_wmma_f32_16x16x4_f32(false, A, false, Bm,
                                                (short)0, acc, false, false);
  }

  float* dst = ws1c + (size_t)qb * (O1 * 25);
  const int mbase = (lane >> 4) * 8;
  for (int r = 0; r < 8; ++r) {
    int m = mbase + r;
    if (m < O1 && col < 25) dst[m * 25 + col] = acc[r];
  }
}

// ---------------------------------------------------------------------------
// Scale-64 y- then z-reductions (tiny; plain VALU).
// ---------------------------------------------------------------------------
__global__ void lncc_s2_y(const float* __restrict__ ws2,    // [qbz][128]
                          float* __restrict__ ws2b) {       // [qbz]
  int i = blockIdx.x * blockDim.x + threadIdx.x;            // 2560
  if (i >= 5 * NB * DIM) return;
  const float* src = ws2 + (size_t)i * DIM;
  float s = 0.f;
#pragma unroll 8
  for (int k = 0; k < K2; ++k) s += src[2 * k];
  ws2b[i] = s;
}

__global__ void lncc_s2_z(const float* __restrict__ ws2b,   // [qb][128]
                          float* __restrict__ ws2c) {       // [qb]
  int i = threadIdx.x;                                      // 20 valid
  if (i >= 5 * NB) return;
  const float* src = ws2b + (size_t)i * DIM;
  float s = 0.f;
#pragma unroll 8
  for (int k = 0; k < K2; ++k) s += src[2 * k];
  ws2c[i] = s;
}

// ---------------------------------------------------------------------------
// Final: per-voxel LNCC, mean, weighted combine.
// ---------------------------------------------------------------------------
__global__ __launch_bounds__(512) void lncc_final(const float* __restrict__ ws1c, // [qb][125]
                                                  const float* __restrict__ ws2c, // [qb]
                                                  float* __restrict__ out) {
  __shared__ float red[512];
  const int tid = threadIdx.x;
  float v = 0.f;
  if (tid < NB * 125) {
    int b = tid / 125;
    int c = tid - b * 125;                     // oz*25 + (oy*5+ox)
    const float numel = 32768.0f;              // 32^3
    float xs = ws1c[(size_t)(0 * NB + b) * 125 + c];
    float ts = ws1c[(size_t)(1 * NB + b) * 125 + c];
    float x2 = ws1c[(size_t)(2 * NB + b) * 125 + c];
    float t2 = ws1c[(size_t)(3 * NB + b) * 125 + c];
    float xt = ws1c[(size_t)(4 * NB + b) * 125 + c];
    float cross = xt - xs * ts / numel;
    float xv = x2 - xs * xs / numel;
    float tv = t2 - ts * ts / numel;
    v = cross * cross / (xv * tv + 1e-5f);
  }
  red[tid] = v;
  __syncthreads();
  for (int s = 256; s > 0; s >>= 1) {
    if (tid < s) red[tid] += red[tid + s];
    __syncthreads();
  }
  if (tid == 0) {
    float sum1 = red[0];
    float sum2 = 0.f;
    const float numel = 262144.0f;             // 64^3
    for (int b = 0; b < NB; ++b) {
      float xs = ws2c[0 * NB + b];
      float ts = ws2c[1 * NB + b];
      float x2 = ws2c[2 * NB + b];
      float t2 = ws2c[3 * NB + b];
      float xt = ws2c[4 * NB + b];
      float cross = xt - xs * ts / numel;
      float xv = x2 - xs * xs / numel;
      float tv = t2 - ts * ts / numel;
      sum2 += cross * cross / (xv * tv + 1e-5f);
    }
    out[0] = 0.4f * (1.0f - sum1 / 500.0f) + 0.6f * (1.0f - sum2 / 4.0f);
  }
}

// ---------------------------------------------------------------------------
extern "C" void kernel_launch(void* const* d_in, const int* in_sizes, int n_in,
                              void* d_out, int out_size, void* d_ws, size_t ws_size,
                              hipStream_t stream) {
  const float* x = (const float*)d_in[0];
  const float* t = (const float*)d_in[1];
  float* out = (float*)d_out;

  // workspace layout (floats); total ~2.04M floats (~8.2 MB)
  float* ws   = (float*)d_ws;
  float* ws1  = ws;                               // 5*4*128*128*5 = 1,638,400
  float* ws2  = ws1 + (size_t)5 * NB * DIM * DIM * O1;   // 5*4*128*128 = 327,680
  float* ws1b = ws2 + (size_t)5 * NB * DIM * DIM;        // 2560*25 = 64,000
  float* ws2b = ws1b + (size_t)5 * NB * DIM * 25;        // 2,560
  float* ws1c = ws2b + (size_t)5 * NB * DIM;             // 20*125 = 2,500
  float* ws2c = ws1c + (size_t)5 * NB * 125;             // 20

  // Stage A: 4*128*4 = 2048 waves, 33792 B dynamic LDS each
  lncc_stageA<<<NB * DIM * 4, 32, 2 * 32 * 132 * sizeof(float), stream>>>(x, t, ws1, ws2);
  // Stage B: one wave per (q,b,z)
  lncc_stageB<<<5 * NB * DIM, 32, 0, stream>>>(ws1, ws1b);
  // Stage C: one wave per (q*4+b, col tile)
  lncc_stageC<<<5 * NB * 2, 32, 0, stream>>>(ws1b, ws1c);
  // Scale-64 reductions
  lncc_s2_y<<<(5 * NB * DIM + 255) / 256, 256, 0, stream>>>(ws2, ws2b);
  lncc_s2_z<<<1, 32, 0, stream>>>(ws2b, ws2c);
  // Final combine
  lncc_final<<<1, 512, 0, stream>>>(ws1c, ws2c, out);
}